// SimRel_48335561949951
// MI455X (gfx1250) — compile-verified
//
#include <hip/hip_runtime.h>
#include <hip/hip_bf16.h>
#include <math.h>

typedef __attribute__((ext_vector_type(2))) float v2f;
typedef __attribute__((ext_vector_type(8))) float v8f;

#define D_DIM 256
#define C_DIM 64
#define EPS_F 1e-8f

// ---------------- Kernel 1: per-class norm + inf flag ----------------
__global__ void SimRel_cnorm_kernel(const float* __restrict__ ca,
                                    float* __restrict__ cn,
                                    float* __restrict__ cinf) {
    int c = threadIdx.x;
    if (c >= C_DIM) return;
    float s = 0.0f;
    int has_inf = 0;
    #pragma unroll 8
    for (int d = 0; d < D_DIM; ++d) {
        float v = ca[c * D_DIM + d];
        has_inf |= isinf(v) ? 1 : 0;
        s += v * v;
    }
    cn[c]   = sqrtf(s);
    cinf[c] = has_inf ? 1.0f : 0.0f;
}

// ---------------- Kernel 2: WMMA cosine-similarity GEMM ----------------
// Block: 256 threads (8 waves). Each wave computes a 16(M) x 64(N) slab.
// Grid: 262144 / (8*16) = 2048 blocks. No tails.
__global__ void __launch_bounds__(256, 2)
SimRel_cossim_kernel(const float* __restrict__ x,
                     const float* __restrict__ ca,
                     const float* __restrict__ cn,
                     const float* __restrict__ cinf,
                     float* __restrict__ out) {
    // B fragments pre-swizzled: sb[kk*256 + t*64 + lane*2 + j]
    //   = class_avgs[t*16 + (lane&15)][4*kk + 2*(lane>>4) + j]
    __shared__ float sb[64 * 4 * 64];   // 16384 floats = 64 KB

    const int tid = threadIdx.x;

    // ---- Stage class_avgs into WMMA B-fragment layout ----
    // Thread tid owns feature d = tid for every class; global reads coalesced.
    {
        const int d   = tid;            // 0..255
        const int kk  = d >> 2;         // k-step
        const int r   = d & 3;          // position within K=4
        const int lhb = (r >> 1) << 4;  // lane-half base (0 or 16)
        const int j   = r & 1;          // float2 element
        for (int c = 0; c < C_DIM; ++c) {
            const int t = c >> 4;       // N tile
            const int n = c & 15;       // column within tile
            sb[(kk * 4 + t) * 64 + (n + lhb) * 2 + j] = ca[c * D_DIM + d];
        }
    }
    __syncthreads();

    const int lane = tid & 31;
    const int wave = tid >> 5;
    const int half = lane >> 4;                         // 0 or 1 (K half)
    const int m0   = blockIdx.x * 128 + wave * 16;      // M tile base
    const int mrow = m0 + (lane & 15);                  // row this lane loads

    const float* arow = x + (size_t)mrow * D_DIM + half * 2;

    v8f acc[4] = {v8f{}, v8f{}, v8f{}, v8f{}};
    float sumsq = 0.0f;

    #pragma unroll 4
    for (int kk = 0; kk < 64; ++kk) {
        // A fragment: 16x4 fp32 tile of inputs (one global_load_b64 / lane)
        v2f a = *(const v2f*)(arow + kk * 4);
        sumsq = fmaf(a.x, a.x, sumsq);
        sumsq = fmaf(a.y, a.y, sumsq);

        // B fragments from LDS: 4 conflict-free ds_load_b64
        const float* bb = &sb[kk * 256 + lane * 2];
        v2f b0 = *(const v2f*)(bb +   0);
        v2f b1 = *(const v2f*)(bb +  64);
        v2f b2 = *(const v2f*)(bb + 128);
        v2f b3 = *(const v2f*)(bb + 192);

        acc[0] = __builtin_amdgcn_wmma_f32_16x16x4_f32(false, a, false, b0,
                                                       (short)0, acc[0], false, false);
        acc[1] = __builtin_amdgcn_wmma_f32_16x16x4_f32(false, a, false, b1,
                                                       (short)0, acc[1], false, false);
        acc[2] = __builtin_amdgcn_wmma_f32_16x16x4_f32(false, a, false, b2,
                                                       (short)0, acc[2], false, false);
        acc[3] = __builtin_amdgcn_wmma_f32_16x16x4_f32(false, a, false, b3,
                                                       (short)0, acc[3], false, false);
    }

    // Row sum-of-squares: lane l and lane l^16 each hold half of row m0+(l&15).
    const float rowsq = sumsq + __shfl_xor(sumsq, 16, 32);

    // Redistribute to C/D layout: VGPR r holds row (r + half*8).
    float rn[8];
    #pragma unroll
    for (int r = 0; r < 8; ++r)
        rn[r] = sqrtf(__shfl(rowsq, (half ? 8 : 0) + r, 32));

    // ---- Scale + store ----
    const int ncol = lane & 15;
    #pragma unroll
    for (int t = 0; t < 4; ++t) {
        const int col   = t * 16 + ncol;
        const float c_n = cn[col];
        const bool  c_i = cinf[col] > 0.0f;
        #pragma unroll
        for (int r = 0; r < 8; ++r) {
            const int row     = m0 + r + half * 8;
            const float dot   = acc[t][r];
            const float denom = fmaxf(rn[r] * c_n, EPS_F);
            const float v     = c_i ? 1.0f : dot / denom;
            out[(size_t)row * C_DIM + col] = v;
        }
    }
}

// ---------------- Host launcher ----------------
extern "C" void kernel_launch(void* const* d_in, const int* in_sizes, int n_in,
                              void* d_out, int out_size, void* d_ws, size_t ws_size,
                              hipStream_t stream) {
    const float* x  = (const float*)d_in[0];   // inputs   (262144, 256) fp32
    // d_in[1] = labels (unused by the reference computation)
    const float* ca = (const float*)d_in[2];   // class_avgs (64, 256) fp32
    float* out = (float*)d_out;                // (262144, 64) fp32

    float* cn   = (float*)d_ws;                // 64 floats
    float* cinf = cn + C_DIM;                  // 64 floats

    SimRel_cnorm_kernel<<<1, 64, 0, stream>>>(ca, cn, cinf);

    const int rows_per_block = 128;            // 8 waves * 16 rows
    const int grid = 262144 / rows_per_block;  // 2048
    SimRel_cossim_kernel<<<grid, 256, 0, stream>>>(x, ca, cn, cinf, out);
}